// SSDLoss_10617159156030
// MI455X (gfx1250) — compile-verified
//
#include <hip/hip_runtime.h>
#include <math.h>

// SSD anchor-matching loss for MI455X (gfx1250, wave32).
// Memory-bound masked reduction (no matmul structure -> no WMMA).
// Uses gfx1250 async global->LDS copy for the entity table when available.

#define IGNORE_TAG_O 1

#ifndef __has_builtin
#define __has_builtin(x) 0
#endif

#if __has_builtin(__builtin_amdgcn_global_load_async_to_lds_b32) && \
    __has_builtin(__builtin_amdgcn_s_wait_asynccnt)
#define USE_ASYNC_LDS 1
#else
#define USE_ASYNC_LDS 0
#endif

typedef __attribute__((address_space(1))) int g1_int;
typedef __attribute__((address_space(3))) int l3_int;

static __device__ __forceinline__ g1_int* to_global_as(const void* p) {
    // global AS shares address values with generic
    return (g1_int*)(unsigned long long)p;
}
static __device__ __forceinline__ l3_int* to_lds_as(const void* p) {
    // generic LDS address: low 32 bits are the LDS offset (ISA 10.2 aperture rules)
    return (l3_int*)(unsigned int)(unsigned long long)p;
}

#if USE_ASYNC_LDS
static __device__ __forceinline__ void async_copy_b32(const void* g, void* l) {
    __builtin_amdgcn_global_load_async_to_lds_b32(to_global_as(g), to_lds_as(l), 0, 0);
}
#endif

static __device__ __forceinline__ float wave_red_add(float v) {
#pragma unroll
    for (int o = 16; o > 0; o >>= 1) v += __shfl_down(v, o, 32);
    return v;
}

constexpr int A_DIM = 8;     // anchors per position (fixed by problem setup)
constexpr int L_DIM = 32;    // classes
constexpr int BLK   = 256;   // 8 waves of 32
constexpr int EMAX  = 1024;  // max entities staged in LDS

__global__ __launch_bounds__(BLK) void ssd_loss_main(
    const int*   __restrict__ input_len,   // [B]
    const float* __restrict__ conf_logits, // [B*S*A]
    const float* __restrict__ cls_logits,  // [B*S*A, L]
    const float* __restrict__ reg_logits,  // [B*S*A, 2]
    const float* __restrict__ label,       // [E, 3] = (type, l, r)
    const int*   __restrict__ ent_index,   // [E]
    const float* __restrict__ anchors,     // [S*A, 2] = (center, size)
    const float* __restrict__ thr_pos_p,
    const float* __restrict__ thr_neg_p,
    float*       __restrict__ partials,    // [gridDim.x, 8]
    int N, int SA, int E)
{
    __shared__ float sh_lab[3 * EMAX];
    __shared__ int   sh_idx[EMAX];
    __shared__ float rbuf[5][BLK / 32];

    const int tid = threadIdx.x;
    const int Ec  = (E < EMAX) ? E : EMAX;

    // ---- stage entity table into LDS (async on gfx1250) ----
#if USE_ASYNC_LDS
    for (int j = tid; j < 3 * Ec; j += BLK) {
        async_copy_b32(label + j, &sh_lab[j]);
    }
    for (int j = tid; j < Ec; j += BLK) {
        async_copy_b32(ent_index + j, &sh_idx[j]);
    }
    __builtin_amdgcn_s_wait_asynccnt(0);
    __syncthreads();
#else
    for (int j = tid; j < 3 * Ec; j += BLK) sh_lab[j] = label[j];
    for (int j = tid; j < Ec; j += BLK)     sh_idx[j] = ent_index[j];
    __syncthreads();
#endif

    const float thr_pos = thr_pos_p[0];
    const float thr_neg = thr_neg_p[0];

    const int t = blockIdx.x * BLK + tid;

    float conf_s = 0.0f, cls_s = 0.0f, reg_s = 0.0f, cls_c = 0.0f, pos_c = 0.0f;

    if (t < N) {
        const int b   = t / SA;
        const int rem = t - b * SA;
        const int s   = rem / A_DIM;

        const float2 an  = ((const float2*)anchors)[rem];
        const float  ac  = an.x;
        const float  asz = an.y;
        const float  al  = ac - 0.5f * asz;
        const float  ar  = ac + 0.5f * asz;

        // max / argmax IoU over this batch's entities (branch is wave-uniform in b)
        float best = -1.0f;
        int   arg  = 0;
        bool  has_ent = false;
        for (int e = 0; e < Ec; ++e) {
            if (sh_idx[e] == b) {
                has_ent = true;
                const float ll = sh_lab[3 * e + 1];
                const float lr = sh_lab[3 * e + 2];
                float inter = fminf(lr, ar) - fmaxf(ll, al);
                inter = fmaxf(inter, 0.0f);
                const float uni = (lr - ll) + (ar - al) - inter;
                const float iou = inter / uni;
                if (iou > best) { best = iou; arg = e; }
            }
        }

        const bool valid  = s < input_len[b];
        const bool active = valid && has_ent;
        const bool pos    = active && (best >= thr_pos);
        const bool neg    = active && (best <= thr_neg);

        // conf: mean over ALL anchors of softplus(x) - x*pos
        {
            const float x = conf_logits[t];
            conf_s = fmaxf(x, 0.0f) + log1pf(expf(-fabsf(x))) - (pos ? x : 0.0f);
        }

        // cls: NLL of log_softmax over L at pos/neg anchors
        if (pos || neg) {
            const int target = pos ? (int)sh_lab[3 * arg + 0] : IGNORE_TAG_O;
            const float4* cp = (const float4*)(cls_logits + (size_t)t * L_DIM);
            float4 v[L_DIM / 4];
#pragma unroll
            for (int j = 0; j < L_DIM / 4; ++j) v[j] = cp[j];
            float m = v[0].x;
#pragma unroll
            for (int j = 0; j < L_DIM / 4; ++j)
                m = fmaxf(m, fmaxf(fmaxf(v[j].x, v[j].y), fmaxf(v[j].z, v[j].w)));
            float ssum = 0.0f;
#pragma unroll
            for (int j = 0; j < L_DIM / 4; ++j)
                ssum += expf(v[j].x - m) + expf(v[j].y - m) +
                        expf(v[j].z - m) + expf(v[j].w - m);
            // scalar refetch of target logit (avoids dynamic register indexing -> scratch)
            const float tv = cls_logits[(size_t)t * L_DIM + target];
            cls_s = m + logf(ssum) - tv;
            cls_c = 1.0f;
        }

        // reg: squared error at positive anchors
        if (pos) {
            const float ll = sh_lab[3 * arg + 1];
            const float lr = sh_lab[3 * arg + 2];
            const float lc = 0.5f * (ll + lr);
            const float ls = lr - ll;
            const float off_c = (lc - ac) / asz;
            const float off_s = ls / asz;
            const float2 rg = ((const float2*)reg_logits)[t];
            const float d0 = rg.x - off_c;
            const float d1 = rg.y - off_s;
            reg_s = d0 * d0 + d1 * d1;
            pos_c = 1.0f;
        }
    }

    // ---- deterministic block reduction (shuffle within wave32, then LDS) ----
    float vals[5] = { conf_s, cls_s, reg_s, cls_c, pos_c };
    const int lane = tid & 31;
    const int wid  = tid >> 5;
#pragma unroll
    for (int k = 0; k < 5; ++k) {
        const float r = wave_red_add(vals[k]);
        if (lane == 0) rbuf[k][wid] = r;
    }
    __syncthreads();
    if (tid == 0) {
        float* p = partials + (size_t)blockIdx.x * 8;
#pragma unroll
        for (int k = 0; k < 5; ++k) {
            float sum = 0.0f;
#pragma unroll
            for (int w = 0; w < BLK / 32; ++w) sum += rbuf[k][w];
            p[k] = sum;
        }
    }
}

__global__ __launch_bounds__(256) void ssd_loss_finalize(
    const float* __restrict__ partials, int nblocks,
    float* __restrict__ out, float invN)
{
    __shared__ float sm[5][256];
    const int tid = threadIdx.x;
    float acc[5] = {0.f, 0.f, 0.f, 0.f, 0.f};
    for (int i = tid; i < nblocks; i += 256) {
        const float* p = partials + (size_t)i * 8;
#pragma unroll
        for (int k = 0; k < 5; ++k) acc[k] += p[k];
    }
#pragma unroll
    for (int k = 0; k < 5; ++k) sm[k][tid] = acc[k];
    __syncthreads();
    for (int s = 128; s > 0; s >>= 1) {
        if (tid < s) {
#pragma unroll
            for (int k = 0; k < 5; ++k) sm[k][tid] += sm[k][tid + s];
        }
        __syncthreads();
    }
    if (tid == 0) {
        const float conf = sm[0][0] * invN;
        const float cls  = sm[1][0] / sm[3][0];
        const float reg  = sm[2][0] / (sm[4][0] * 2.0f);
        out[0] = conf + cls + reg;  // W_CONF = W_CLS = W_REG = 1
        out[1] = conf;
        out[2] = cls;
        out[3] = reg;
    }
}

extern "C" void kernel_launch(void* const* d_in, const int* in_sizes, int n_in,
                              void* d_out, int out_size, void* d_ws, size_t ws_size,
                              hipStream_t stream) {
    const int*   input_len   = (const int*)  d_in[0];
    const float* conf_logits = (const float*)d_in[1];
    const float* cls_logits  = (const float*)d_in[2];
    const float* reg_logits  = (const float*)d_in[3];
    const float* label       = (const float*)d_in[4];
    const int*   ent_index   = (const int*)  d_in[5];
    const float* anchors     = (const float*)d_in[6];
    const float* thr_pos     = (const float*)d_in[7];
    const float* thr_neg     = (const float*)d_in[8];

    const int N  = in_sizes[1];          // B*S*A
    const int SA = in_sizes[6] / 2;      // S*A
    const int E  = in_sizes[5];          // entities

    const int nblocks = (N + BLK - 1) / BLK;
    float* partials = (float*)d_ws;      // nblocks * 8 floats

    ssd_loss_main<<<nblocks, BLK, 0, stream>>>(
        input_len, conf_logits, cls_logits, reg_logits,
        label, ent_index, anchors, thr_pos, thr_neg,
        partials, N, SA, E);

    ssd_loss_finalize<<<1, 256, 0, stream>>>(
        partials, nblocks, (float*)d_out, 1.0f / (float)N);
}